// Net2_25348896981188
// MI455X (gfx1250) — compile-verified
//
#include <hip/hip_runtime.h>
#include <math.h>

// ---------------- problem constants (from reference) ----------------
#define BG   128                 // graphs
#define NN   2048                // nodes per graph
#define EE   2097152             // edges
#define M1   (BG * NN)           // 262144 nodes, layer 1
#define KP1  1639                // K1 = ceil(0.8*N)
#define KP2  1312                // K2 = ceil(0.8*K1)
#define M2   (BG * KP1)          // 209792 nodes, layer 2

typedef __attribute__((ext_vector_type(2))) float v2f;
typedef __attribute__((ext_vector_type(8))) float v8f;

// =====================================================================
// Fused node GEMM:  Y = X @ Wrel   (cols 0..31, no bias)
//                   H = X @ Wroot + bias (cols 32..63)
// via V_WMMA_F32_16X16X4_F32.  256 threads = 8 waves, 16 rows per wave,
// 128 rows per block.  K is 64 (conv1) or 32 (conv2).
// =====================================================================
template <int K>
__global__ void gemm_node(const float* __restrict__ X,
                          const float* __restrict__ Wrel,
                          const float* __restrict__ Wroot,
                          const float* __restrict__ bias,
                          float* __restrict__ Y,
                          float* __restrict__ H) {
  __shared__ float ldsW[K * 64];            // concat [Wrel | Wroot], row-major K x 64
  const int tid = threadIdx.x;
  for (int idx = tid; idx < K * 64; idx += 256) {
    const int kk = idx >> 6, n = idx & 63;
    ldsW[idx] = (n < 32) ? Wrel[kk * 32 + n] : Wroot[kk * 32 + (n - 32)];
  }
  __syncthreads();

  const int wave = tid >> 5;
  const int lane = tid & 31;
  const int half = lane >> 4;               // 0: lanes 0-15, 1: lanes 16-31
  const int l15  = lane & 15;
  const size_t rowBase = ((size_t)blockIdx.x * 8 + wave) * 16;

  // A fragment source: row = rowBase + l15, cols (k + 2*half, k + 2*half + 1)
  const float* aptr = X + (rowBase + l15) * K + half * 2;

  v8f acc0 = {0,0,0,0,0,0,0,0};
  v8f acc1 = acc0, acc2 = acc0, acc3 = acc0;

#pragma unroll
  for (int k = 0; k < K; k += 4) {
    const v2f a = *(const v2f*)(aptr + k);                 // A 16x4 frag
    const int kb = (k + half * 2) * 64 + l15;              // B 4x16 frags from LDS
    const v2f b0 = { ldsW[kb +  0], ldsW[kb + 64 +  0] };
    const v2f b1 = { ldsW[kb + 16], ldsW[kb + 64 + 16] };
    const v2f b2 = { ldsW[kb + 32], ldsW[kb + 64 + 32] };
    const v2f b3 = { ldsW[kb + 48], ldsW[kb + 64 + 48] };
    acc0 = __builtin_amdgcn_wmma_f32_16x16x4_f32(false, a, false, b0, (short)0, acc0, false, false);
    acc1 = __builtin_amdgcn_wmma_f32_16x16x4_f32(false, a, false, b1, (short)0, acc1, false, false);
    acc2 = __builtin_amdgcn_wmma_f32_16x16x4_f32(false, a, false, b2, (short)0, acc2, false, false);
    acc3 = __builtin_amdgcn_wmma_f32_16x16x4_f32(false, a, false, b3, (short)0, acc3, false, false);
  }

  // C/D layout: VGPR r -> row (half*8 + r), col l15 within the 16-wide tile
  const size_t row0 = rowBase + half * 8;
  const float bia2 = bias[l15];
  const float bia3 = bias[16 + l15];
#pragma unroll
  for (int r = 0; r < 8; ++r) {
    const size_t ro = (row0 + r) * 32;
    Y[ro + l15]      = acc0[r];
    Y[ro + 16 + l15] = acc1[r];
    H[ro + l15]      = acc2[r] + bia2;
    H[ro + 16 + l15] = acc3[r] + bia3;
  }
}

// =====================================================================
// Edge aggregation:  H[dst] += Y[src] * ew   (32 feats, 8 lanes/edge x f4)
// mapping != nullptr -> remap through TopK selection (skip invalid edges)
// =====================================================================
__global__ void edge_agg(const int* __restrict__ src, const int* __restrict__ dst,
                         const float* __restrict__ ew,
                         const float* __restrict__ Y, float* __restrict__ H,
                         const int* __restrict__ mapping) {
  const unsigned idx = blockIdx.x * blockDim.x + threadIdx.x;
  const unsigned e = idx >> 3;
  const unsigned part = idx & 7;
  if (e >= EE) return;
  int s = src[e], d = dst[e];
  if (mapping) {
    s = mapping[s];
    d = mapping[d];
    if (s < 0 || d < 0) return;
  }
  const float w = ew[e];
  const float4 v = ((const float4*)(Y + (size_t)s * 32))[part];
  float* o = H + (size_t)d * 32 + part * 4;
  atomicAdd(o + 0, v.x * w);
  atomicAdd(o + 1, v.y * w);
  atomicAdd(o + 2, v.z * w);
  atomicAdd(o + 3, v.w * w);
}

// =====================================================================
// In-place ReLU + TopK score:  score = tanh((relu(h) . p) / ||p||)
// =====================================================================
__global__ void relu_score(float* __restrict__ H, float* __restrict__ score,
                           const float* __restrict__ p, int M) {
  const int i = blockIdx.x * blockDim.x + threadIdx.x;
  if (i >= M) return;
  float pn = 0.f, acc = 0.f;
  float* h = H + (size_t)i * 32;
#pragma unroll
  for (int j = 0; j < 32; ++j) {
    const float pj = p[j];
    pn += pj * pj;
    float v = fmaxf(h[j], 0.0f);
    h[j] = v;
    acc += v * pj;
  }
  score[i] = tanhf(acc / sqrtf(pn));
}

// =====================================================================
// TopK pooling: rank-by-count (matches lax.top_k tie-break: value desc,
// index asc).  rank < k -> selected; pooled[g*k+rank] = h * score.
// Optionally writes old->new node mapping (-1 if dropped).
// =====================================================================
__global__ void pool_rank(const float* __restrict__ score, const float* __restrict__ H,
                          float* __restrict__ pooled, int* __restrict__ mapping,
                          int n, int k) {
  __shared__ float s[2048];
  const int g = blockIdx.x;
  const int base = g * n;
  for (int i = threadIdx.x; i < n; i += blockDim.x) s[i] = score[base + i];
  __syncthreads();
  for (int i = threadIdx.x; i < n; i += blockDim.x) {
    const float si = s[i];
    int rank = 0;
    for (int j = 0; j < n; ++j) {
      const float sj = s[j];
      rank += (sj > si) || (sj == si && j < i);
    }
    const int v = base + i;
    if (rank < k) {
      const int nw = g * k + rank;
      if (mapping) mapping[v] = nw;
      const float4* hs = (const float4*)(H + (size_t)v * 32);
      float4* od = (float4*)(pooled + (size_t)nw * 32);
#pragma unroll
      for (int t = 0; t < 8; ++t) {
        float4 q = hs[t];
        q.x *= si; q.y *= si; q.z *= si; q.w *= si;
        od[t] = q;
      }
    } else if (mapping) {
      mapping[v] = -1;
    }
  }
}

// =====================================================================
// Graph readout:  out[g] = concat(max_r Xp[g,r,:], mean_r Xp[g,r,:])  [64]
// =====================================================================
__global__ void pool_readout(const float* __restrict__ Xp, float* __restrict__ out,
                             int rows) {
  __shared__ float smx[256], ssm[256];
  const int g = blockIdx.x;
  const int tid = threadIdx.x;
  const int d = tid & 31, grp = tid >> 5;              // 8 row-groups x 32 dims
  float mx = -3.402823466e38f, sm = 0.f;
  for (int r = grp; r < rows; r += 8) {
    const float v = Xp[((size_t)g * rows + r) * 32 + d];
    mx = fmaxf(mx, v);
    sm += v;
  }
  smx[tid] = mx; ssm[tid] = sm;
  __syncthreads();
  if (tid < 32) {
    for (int q = 1; q < 8; ++q) {
      mx = fmaxf(mx, smx[q * 32 + tid]);
      sm += ssm[q * 32 + tid];
    }
    out[g * 64 + tid]      = mx;
    out[g * 64 + 32 + tid] = sm / (float)rows;
  }
}

// =====================================================================
// Readout MLP + log_softmax: one thread per graph.
// =====================================================================
__global__ void readout_mlp(const float* __restrict__ x1, const float* __restrict__ x2,
                            const float* __restrict__ W1, const float* __restrict__ b1,
                            const float* __restrict__ W2, const float* __restrict__ b2,
                            const float* __restrict__ W3, const float* __restrict__ b3,
                            float* __restrict__ out) {
  const int g = blockIdx.x * blockDim.x + threadIdx.x;
  if (g >= BG) return;
  float z[64];
#pragma unroll
  for (int i = 0; i < 64; ++i) z[i] = x1[g * 64 + i] + x2[g * 64 + i];
  float z1[32];
#pragma unroll
  for (int o = 0; o < 32; ++o) {
    float a = b1[o];
#pragma unroll
    for (int i = 0; i < 64; ++i) a += z[i] * W1[i * 32 + o];
    z1[o] = fmaxf(a, 0.f);
  }
  float z2[16];
#pragma unroll
  for (int o = 0; o < 16; ++o) {
    float a = b2[o];
#pragma unroll
    for (int i = 0; i < 32; ++i) a += z1[i] * W2[i * 16 + o];
    z2[o] = fmaxf(a, 0.f);
  }
  float l0 = b3[0], l1 = b3[1];
#pragma unroll
  for (int i = 0; i < 16; ++i) {
    l0 += z2[i] * W3[i * 2 + 0];
    l1 += z2[i] * W3[i * 2 + 1];
  }
  const float m = fmaxf(l0, l1);
  const float lse = m + logf(expf(l0 - m) + expf(l1 - m));
  out[g * 2 + 0] = l0 - lse;
  out[g * 2 + 1] = l1 - lse;
}

// =====================================================================
extern "C" void kernel_launch(void* const* d_in, const int* in_sizes, int n_in,
                              void* d_out, int out_size, void* d_ws, size_t ws_size,
                              hipStream_t stream) {
  (void)in_sizes; (void)n_in; (void)out_size; (void)ws_size;
  const float* x   = (const float*)d_in[0];
  const int*   ei  = (const int*)d_in[1];       // [2, E] int32
  const float* ea  = (const float*)d_in[2];
  /* d_in[3] = batch (unused: graphs are contiguous blocks) */
  const float* W1r = (const float*)d_in[4];
  const float* b1  = (const float*)d_in[5];
  const float* W1o = (const float*)d_in[6];
  const float* p1  = (const float*)d_in[7];
  const float* W2r = (const float*)d_in[8];
  const float* b2  = (const float*)d_in[9];
  const float* W2o = (const float*)d_in[10];
  const float* p2  = (const float*)d_in[11];
  const float* L1W = (const float*)d_in[12];
  const float* L1b = (const float*)d_in[13];
  const float* L2W = (const float*)d_in[14];
  const float* L2b = (const float*)d_in[15];
  const float* L3W = (const float*)d_in[16];
  const float* L3b = (const float*)d_in[17];
  const int* src = ei;
  const int* dst = ei + EE;

  // ---- workspace layout (with reuse; all buffers fully written before read)
  char* ws = (char*)d_ws;
  const size_t SZ_A = (size_t)M1 * 32 * 4;           // 33.55 MB
  const size_t SZ_B = (size_t)M1 * 32 * 4;           // 33.55 MB
  const size_t SZ_C = (size_t)M2 * 32 * 4;           // 26.85 MB
  const size_t SZ_D = (size_t)M2 * 32 * 4;           // 26.85 MB
  float* Ybuf  = (float*)(ws);                       // conv1 y; later reused as H3
  float* Hbuf  = (float*)(ws + SZ_A);                // conv1 h; later reused as Y2
  float* H2in  = (float*)(ws + SZ_A + SZ_B);         // pooled layer-1 output
  float* H2buf = (float*)(ws + SZ_A + SZ_B + SZ_C);  // conv2 h
  char* small  = ws + SZ_A + SZ_B + SZ_C + SZ_D;
  float* score1  = (float*)(small);                         small += (size_t)M1 * 4;
  int*   mapping = (int*)(small);                           small += (size_t)M1 * 4;
  float* score2  = (float*)(small);                         small += (size_t)M2 * 4;
  float* x1      = (float*)(small);                         small += (size_t)BG * 64 * 4;
  float* x2      = (float*)(small);
  float* Y2 = Hbuf;   // Hbuf dead after pool1
  float* H3 = Ybuf;   // Ybuf dead after conv1 edge pass

  // ---------------- conv1: GEMM (WMMA) + edge aggregation + relu/score
  gemm_node<64><<<M1 / 128, 256, 0, stream>>>(x, W1r, W1o, b1, Ybuf, Hbuf);
  edge_agg<<<(EE * 8) / 256, 256, 0, stream>>>(src, dst, ea, Ybuf, Hbuf, nullptr);
  relu_score<<<(M1 + 255) / 256, 256, 0, stream>>>(Hbuf, score1, p1, M1);

  // ---------------- pool1 (TopK K1) + graph readout x1
  pool_rank<<<BG, 512, 0, stream>>>(score1, Hbuf, H2in, mapping, NN, KP1);
  pool_readout<<<BG, 256, 0, stream>>>(H2in, x1, KP1);

  // ---------------- conv2: GEMM (WMMA) + remapped edge aggregation
  gemm_node<32><<<M2 / 128, 256, 0, stream>>>(H2in, W2r, W2o, b2, Y2, H2buf);
  edge_agg<<<(EE * 8) / 256, 256, 0, stream>>>(src, dst, ea, Y2, H2buf, mapping);
  relu_score<<<(M2 + 255) / 256, 256, 0, stream>>>(H2buf, score2, p2, M2);

  // ---------------- pool2 (TopK K2) + graph readout x2
  pool_rank<<<BG, 512, 0, stream>>>(score2, H2buf, H3, nullptr, KP1, KP2);
  pool_readout<<<BG, 256, 0, stream>>>(H3, x2, KP2);

  // ---------------- readout MLP + log_softmax
  readout_mlp<<<1, 128, 0, stream>>>(x1, x2, L1W, L1b, L2W, L2b, L3W, L3b,
                                     (float*)d_out);
}